// EmbeddingToExpression_45157286150935
// MI455X (gfx1250) — compile-verified
//
#include <hip/hip_runtime.h>
#include <hip/hip_bf16.h>
#include <stdint.h>

// Problem constants (match reference)
#define CELLS 4096
#define REGB  4096
#define NE 5
#define NI 5

// Tiling
#define BTILE 256                     // regions per block == blockDim.x
#define ATILE 32                      // cell rows per block
#define ROWF (BTILE * NE)             // 1280 floats staged per row
#define ROWB (ROWF * 4)               // 5120 bytes per staged row
#define GROWB (REGB * NE * 4)         // 81920 bytes: global stride between cell rows

// Pipelining: 4-row chunks, 4 LDS buffers, 5 async B128 per thread per chunk
#define CHUNK_ROWS 4
#define CHUNKF (CHUNK_ROWS * ROWF)    // 5120 floats per chunk
#define CHUNKB (CHUNK_ROWS * ROWB)    // 20480 bytes per chunk
#define NBUF 4
#define OPS 5                         // (CHUNKB/16)/BTILE == 1280/256
#define NCHUNKS (ATILE / CHUNK_ROWS)  // 8

// ---- CDNA5 async global->LDS path -----------------------------------------
#if defined(__HIP_DEVICE_COMPILE__) && __has_builtin(__builtin_amdgcn_global_load_async_to_lds_b128)
#define USE_ASYNC 1
#else
#define USE_ASYNC 0
#endif

// Builtin params are int4* in AS1 (global) / AS3 (LDS).
typedef int v4i_t __attribute__((__vector_size__(4 * sizeof(int))));
typedef __attribute__((address_space(1))) v4i_t as1_v4i;
typedef __attribute__((address_space(3))) v4i_t as3_v4i;

#define AS1P(p) ((as1_v4i*)(uintptr_t)(p))
#define AS3P(p) ((as3_v4i*)(uint32_t)(uintptr_t)(p))

#if USE_ASYNC
#if __has_builtin(__builtin_amdgcn_s_wait_asynccnt)
#define WAIT_ASYNC(n) __builtin_amdgcn_s_wait_asynccnt(n)
#else
#define WAIT_ASYNC(n) asm volatile("s_wait_asynccnt %0" :: "i"(n) : "memory")
#endif
#else
#define WAIT_ASYNC(n) ((void)0)
#endif

__global__ __launch_bounds__(BTILE, 2)
void EmbeddingToExpression_kernel(const float* __restrict__ emb,
                                  const int*   __restrict__ region_ix,
                                  const float* __restrict__ w1g,   // [20000,5,5]
                                  const float* __restrict__ b1g,   // [20000,5]
                                  const float* __restrict__ w2g,   // [20000,5]
                                  const float* __restrict__ b2g,   // [20000]
                                  float* __restrict__ out)
{
    __shared__ __align__(16) float smem[NBUF * CHUNKF];   // 80 KB

    const int t  = threadIdx.x;
    const int b0 = blockIdx.x * BTILE;        // region tile
    const int a0 = blockIdx.y * ATILE;        // cell tile
    const int b  = b0 + t;

    // ---- loop-invariant decomposition of this thread's 5 chunk transfers ----
    // chunk-local 16B-chunk id q = t + i*256 -> row q/320, byte offset (q%320)*16
    int grow[OPS];   // byte offset into global chunk (row*GROWB + inrow*16)
    int soff[OPS];   // byte offset into LDS buffer (q*16)
#pragma unroll
    for (int i = 0; i < OPS; ++i) {
        const int q = t + i * BTILE;
        grow[i] = (q / 320) * GROWB + (q % 320) * 16;
        soff[i] = q * 16;
    }

    const char* gBase = (const char*)(emb + (size_t)a0 * (REGB * NE) + (size_t)b0 * NE);

    // stage chunk k into LDS buffer k%NBUF (5 async B128 per thread, uniform per wave)
    auto stage = [&](int k) {
        const char* gc = gBase + (size_t)k * (CHUNK_ROWS * (size_t)GROWB);
        char*       sc = ((char*)smem) + (k % NBUF) * CHUNKB;
#if USE_ASYNC
#pragma unroll
        for (int i = 0; i < OPS; ++i)
            __builtin_amdgcn_global_load_async_to_lds_b128(
                AS1P(gc + grow[i]), AS3P(sc + soff[i]), 0, 0);
#else
#pragma unroll
        for (int i = 0; i < OPS; ++i)
            *(float4*)(sc + soff[i]) = *(const float4*)(gc + grow[i]);
#endif
    };

    // ---- launch the HBM stream FIRST: NBUF-1 chunks (15 B128/wave) in flight
    // so the scattered parameter gather below overlaps the stream's latency.
#pragma unroll
    for (int k = 0; k < NBUF - 1; ++k) stage(k);        // chunks 0,1,2

    // ---- gather per-region MLP parameters into registers (L2-resident) ----
    const int r = region_ix[b];
    float w1[NE][NI];
    const float* w1p = w1g + (size_t)r * (NE * NI);
#pragma unroll
    for (int c = 0; c < NE; ++c)
#pragma unroll
        for (int d = 0; d < NI; ++d)
            w1[c][d] = w1p[c * NI + d];

    float b1[NI], w2[NI];
    const float* b1p = b1g + (size_t)r * NI;
    const float* w2p = w2g + (size_t)r * NI;
#pragma unroll
    for (int d = 0; d < NI; ++d) { b1[d] = b1p[d]; w2[d] = w2p[d]; }
    const float b2 = b2g[r];

    // ---- software-pipelined stream: retire oldest chunk, refill, compute ----
    for (int k = 0; k < NCHUNKS; ++k) {
        // retire chunk k: allow (remaining issued beyond k) * OPS to stay in flight
        const int rem = NCHUNKS - 1 - k;
        if (rem >= NBUF - 2)      WAIT_ASYNC((NBUF - 2) * OPS);  // 10
        else if (rem == 1)        WAIT_ASYNC(OPS);               // 5
        else                      WAIT_ASYNC(0);
        __syncthreads();          // chunk k visible to all; buffer (k-1)%NBUF free

        if (k + NBUF - 1 < NCHUNKS) stage(k + NBUF - 1);

        const float* buf = smem + (k % NBUF) * CHUNKF;
#pragma unroll
        for (int j = 0; j < CHUNK_ROWS; ++j) {
            const float* cur = buf + j * ROWF;

            float e[NE];
#pragma unroll
            for (int c = 0; c < NE; ++c) e[c] = cur[t * NE + c];

            float acc[NI];
#pragma unroll
            for (int d = 0; d < NI; ++d) acc[d] = b1[d];
#pragma unroll
            for (int c = 0; c < NE; ++c)
#pragma unroll
                for (int d = 0; d < NI; ++d)
                    acc[d] = fmaf(e[c], w1[c][d], acc[d]);

            float o = b2;
#pragma unroll
            for (int d = 0; d < NI; ++d) {
                const float h = 1.0f / (1.0f + __expf(-acc[d]));   // sigmoid via v_exp_f32
                o = fmaf(h, w2[d], o);
            }

            __builtin_nontemporal_store(
                o, &out[(size_t)(a0 + k * CHUNK_ROWS + j) * REGB + b]);
        }
    }
}

extern "C" void kernel_launch(void* const* d_in, const int* in_sizes, int n_in,
                              void* d_out, int out_size, void* d_ws, size_t ws_size,
                              hipStream_t stream) {
    (void)in_sizes; (void)n_in; (void)out_size; (void)d_ws; (void)ws_size;
    const float* emb = (const float*)d_in[0];
    const int*   rix = (const int*)  d_in[1];
    const float* w1  = (const float*)d_in[2];
    const float* b1  = (const float*)d_in[3];
    const float* w2  = (const float*)d_in[4];
    const float* b2  = (const float*)d_in[5];
    float* out = (float*)d_out;

    dim3 grid(REGB / BTILE, CELLS / ATILE);   // 16 x 128 = 2048 workgroups
    EmbeddingToExpression_kernel<<<grid, BTILE, 0, stream>>>(emb, rix, w1, b1, w2, b2, out);
}